// AblationGCN_56521769616161
// MI455X (gfx1250) — compile-verified
//
#include <hip/hip_runtime.h>

// ---------------------------------------------------------------------------
// AblationGCN on MI455X (gfx1250, wave32).
// Memory-bound gather/scatter GCN; LayerNorm reductions done on the matrix
// pipes with V_WMMA_F32_16X16X4_F32 (B = ones -> rowwise sums in f32).
// Scatter-add pinned to hardware global_atomic_add_f32 via inline asm.
// ---------------------------------------------------------------------------

#define DFEAT   128
#define LN_EPS  1e-5f

typedef __attribute__((ext_vector_type(2))) float v2f;
typedef __attribute__((ext_vector_type(8))) float v8f;

__device__ __forceinline__ float frelu(float x) { return fmaxf(x, 0.0f); }

// Guaranteed single-instruction, no-return f32 atomic add at device scope.
// The RMW executes in the L2 atomic units (agg is L2-resident: 51 MB < 192 MB).
__device__ __forceinline__ void atomic_add_f32(float* p, float v) {
  asm volatile("global_atomic_add_f32 %0, %1, off scope:SCOPE_DEV"
               :: "v"(p), "v"(v)
               : "memory");
}

// Select v[i], i in [0,8), without scratch: 3-level cndmask tree.
__device__ __forceinline__ float sel8(v8f v, int i) {
  float x01 = (i & 1) ? v[1] : v[0];
  float x23 = (i & 1) ? v[3] : v[2];
  float x45 = (i & 1) ? v[5] : v[4];
  float x67 = (i & 1) ? v[7] : v[6];
  float a   = (i & 2) ? x23 : x01;
  float b   = (i & 2) ? x67 : x45;
  return (i & 4) ? b : a;
}

// ---------------------------------------------------------------------------
// Utility kernels
// ---------------------------------------------------------------------------
__global__ void gcn_fill_f32(float* __restrict__ p, float val, int n) {
  int i = blockIdx.x * blockDim.x + threadIdx.x;
  if (i < n) p[i] = val;
}

__global__ void gcn_zero4(float4* __restrict__ p, int n4) {
  int i = blockIdx.x * blockDim.x + threadIdx.x;
  if (i < n4) p[i] = make_float4(0.f, 0.f, 0.f, 0.f);
}

// deg[row[e]] += 1  (deg pre-filled with 1.0 for the self contribution)
__global__ void gcn_deg(float* __restrict__ deg, const int* __restrict__ row, int E) {
  int e = blockIdx.x * blockDim.x + threadIdx.x;
  if (e < E) atomic_add_f32(&deg[row[e]], 1.0f);
}

__global__ void gcn_dis(float* __restrict__ dis, const float* __restrict__ deg, int n) {
  int i = blockIdx.x * blockDim.x + threadIdx.x;
  if (i < n) dis[i] = rsqrtf(deg[i]);
}

// ---------------------------------------------------------------------------
// Edge kernel: one wave per edge. Lane l owns features [4l, 4l+4).
//   agg[col] += dis[row]*dis[col] * relu(h[row])
// h (51 MB) is L2-resident (192 MB L2) -> gathers and atomics stay in L2.
// ---------------------------------------------------------------------------
__global__ void __launch_bounds__(256)
gcn_edge(const float* __restrict__ h, float* __restrict__ agg,
         const float* __restrict__ dis, const int* __restrict__ row,
         const int* __restrict__ col, int E) {
  int w    = (int)((blockIdx.x * (unsigned)blockDim.x + threadIdx.x) >> 5);
  int lane = threadIdx.x & 31;
  if (w >= E) return;                      // wave-uniform exit

  int r = row[w];
  int c = col[w];
  if (w + 8 < E) {                         // hint next edge's source row into cache
    __builtin_prefetch(h + (size_t)row[w + 8] * DFEAT + lane * 4, 0, 1);
  }
  float norm = dis[r] * dis[c];

  const float4* hr = (const float4*)(h + (size_t)r * DFEAT);
  float4 x = hr[lane];                     // global_load_b128

  float* dst = agg + (size_t)c * DFEAT + lane * 4;
  atomic_add_f32(dst + 0, frelu(x.x) * norm);
  atomic_add_f32(dst + 1, frelu(x.y) * norm);
  atomic_add_f32(dst + 2, frelu(x.z) * norm);
  atomic_add_f32(dst + 3, frelu(x.w) * norm);
}

// ---------------------------------------------------------------------------
// Fused node kernel: v = agg + relu(h + root)/deg, then LayerNorm (+ReLU).
// One wave handles 16 node rows. Rowwise sum(v) and sum(v^2) are computed with
// V_WMMA_F32_16X16X4_F32 against a ones B-matrix:
//   A layout (16x4 f32): lanes 0-15 hold K={0,1}, lanes 16-31 hold K={2,3},
//   so lane L feeds features 4k + 2*(L>>4) + {0,1} of node (L&15).
//   D[m][n] = rowsum(m) for all n; extracted via sel8 + ds_bpermute_b32.
// ---------------------------------------------------------------------------
__global__ void __launch_bounds__(256)
gcn_node(const float* __restrict__ agg, const float* __restrict__ h,
         const float* __restrict__ root, const float* __restrict__ gamma,
         const float* __restrict__ beta, const float* __restrict__ deg,
         float* __restrict__ out, int nNodes, int doRelu) {
  __shared__ __align__(16) float sroot[DFEAT];
  __shared__ __align__(16) float sg[DFEAT];
  __shared__ __align__(16) float sb[DFEAT];

  int t = threadIdx.x;
  if (t < DFEAT) {
    sroot[t] = root[t];
    sg[t]    = gamma[t];
    sb[t]    = beta[t];
  }
  __syncthreads();

  int lane = t & 31;
  int tile = blockIdx.x * 8 + (t >> 5);    // 16 nodes per wave
  if (tile * 16 >= nNodes) return;         // wave-uniform: EXEC stays all-1s

  int m    = lane & 15;
  int half = lane >> 4;
  int node = tile * 16 + m;
  if (node >= nNodes) node = nNodes - 1;   // benign duplicate on ragged tail

  float dInv = 1.0f / deg[node];
  const float* aRow = agg + (size_t)node * DFEAT;
  const float* hRow = h   + (size_t)node * DFEAT;

  v2f ones; ones[0] = 1.0f; ones[1] = 1.0f;
  v8f s1 = {};                             // sum(v)
  v8f s2 = {};                             // sum(v^2)
  v2f va[32];                              // this lane's 64 features of node

#pragma unroll
  for (int k = 0; k < 32; ++k) {
    int f = 4 * k + 2 * half;
    float2 a  = *(const float2*)(aRow + f);
    float2 hh = *(const float2*)(hRow + f);
    float2 rr = *(const float2*)(&sroot[f]);           // ds_load_b64
    float vx = a.x + frelu(hh.x + rr.x) * dInv;
    float vy = a.y + frelu(hh.y + rr.y) * dInv;
    v2f av;  av[0]  = vx;       av[1]  = vy;
    v2f av2; av2[0] = vx * vx;  av2[1] = vy * vy;
    va[k] = av;
    s1 = __builtin_amdgcn_wmma_f32_16x16x4_f32(false, av,  false, ones, (short)0, s1, false, false);
    s2 = __builtin_amdgcn_wmma_f32_16x16x4_f32(false, av2, false, ones, (short)0, s2, false, false);
  }

  // Extract this node's sums: local 8-way select, then cross-half bpermute.
  int i3      = lane & 7;
  int srcLane = (lane & 7) | ((lane & 8) << 1);
  float S1 = __builtin_bit_cast(float,
      __builtin_amdgcn_ds_bpermute(srcLane << 2, __builtin_bit_cast(int, sel8(s1, i3))));
  float S2 = __builtin_bit_cast(float,
      __builtin_amdgcn_ds_bpermute(srcLane << 2, __builtin_bit_cast(int, sel8(s2, i3))));

  float mu   = S1 * (1.0f / DFEAT);
  float var  = S2 * (1.0f / DFEAT) - mu * mu;
  float rstd = rsqrtf(var + LN_EPS);

  float* oRow = out + (size_t)node * DFEAT;
#pragma unroll
  for (int k = 0; k < 32; ++k) {
    int f = 4 * k + 2 * half;
    float2 gg = *(const float2*)(&sg[f]);
    float2 bb = *(const float2*)(&sb[f]);
    float ox = (va[k][0] - mu) * rstd * gg.x + bb.x;
    float oy = (va[k][1] - mu) * rstd * gg.y + bb.y;
    if (doRelu) { ox = frelu(ox); oy = frelu(oy); }
    float2 o; o.x = ox; o.y = oy;
    *(float2*)(oRow + f) = o;
  }
}

// ---------------------------------------------------------------------------
// Host launcher
// ---------------------------------------------------------------------------
extern "C" void kernel_launch(void* const* d_in, const int* in_sizes, int n_in,
                              void* d_out, int out_size, void* d_ws, size_t ws_size,
                              hipStream_t stream) {
  const float* in_feat = (const float*)d_in[0];
  const int*   ei      = (const int*)d_in[1];   // [2,E] row-major: row then col
  const float* root0   = (const float*)d_in[2];
  const float* root1   = (const float*)d_in[3];
  const float* g0      = (const float*)d_in[4];
  const float* b0      = (const float*)d_in[5];
  const float* g1      = (const float*)d_in[6];
  const float* b1      = (const float*)d_in[7];
  float*       out     = (float*)d_out;

  const int N = in_sizes[0] / DFEAT;
  const int E = in_sizes[1] / 2;
  const int* row = ei;
  const int* col = ei + E;

  float* deg = (float*)d_ws;       // N floats
  float* dis = deg + N;            // N floats
  float* A   = dis + N;            // N*DFEAT floats (scatter-add target)
  const int ND4 = (N * DFEAT) / 4;

  const int thr = 256;
  int edgeBlocks = (int)(((long long)E * 32 + thr - 1) / thr);   // 1 wave/edge
  int nodeBlocks = (((N + 15) / 16) + 7) / 8;                    // 8 waves/block

  // Degrees: deg = 1 + out-degree(row);  dis = deg^-1/2
  gcn_fill_f32<<<(N + thr - 1) / thr, thr, 0, stream>>>(deg, 1.0f, N);
  gcn_deg<<<(E + thr - 1) / thr, thr, 0, stream>>>(deg, row, E);
  gcn_dis<<<(N + thr - 1) / thr, thr, 0, stream>>>(dis, deg, N);

  // Layer 0: agg -> root+LN+ReLU  (h1 stored in d_out)
  gcn_zero4<<<(ND4 + thr - 1) / thr, thr, 0, stream>>>((float4*)A, ND4);
  gcn_edge<<<edgeBlocks, thr, 0, stream>>>(in_feat, A, dis, row, col, E);
  gcn_node<<<nodeBlocks, thr, 0, stream>>>(A, in_feat, root0, g0, b0, deg, out, N, 1);

  // Layer 1: agg -> root+LN  (reads h1 from d_out, writes final in-place)
  gcn_zero4<<<(ND4 + thr - 1) / thr, thr, 0, stream>>>((float4*)A, ND4);
  gcn_edge<<<edgeBlocks, thr, 0, stream>>>(out, A, dis, row, col, E);
  gcn_node<<<nodeBlocks, thr, 0, stream>>>(A, out, root1, g1, b1, deg, out, N, 0);
}